// MultiBoxLoss_54666343744057
// MI455X (gfx1250) — compile-verified
//
#include <hip/hip_runtime.h>
#include <math.h>

#define BN   64
#define PN   8732
#define NC   91
#define NOBJn 16
#define IMGf 300.0f
#define IOUTH 0.5f

typedef __attribute__((ext_vector_type(2))) float v2f;
typedef __attribute__((ext_vector_type(8))) float v8f;
typedef int v4i_gcc __attribute__((vector_size(16)));
typedef __attribute__((address_space(1))) v4i_gcc as1_v4i;
typedef __attribute__((address_space(3))) v4i_gcc as3_v4i;

// ---------------- K0: zero accumulators ----------------
__global__ void k_init(unsigned long long* bestkey, int* numpos, float* scal) {
  int i = blockIdx.x * 256 + threadIdx.x;
  if (i < BN * NOBJn) bestkey[i] = 0ull;
  if (i < BN) numpos[i] = 0;
  if (i < 2) scal[i] = 0.0f;
}

// ---------------- K1: match (IoU, per-prior best GT, per-GT best prior) ----------------
__global__ void k_match(const float* __restrict__ gt_boxes, const float* __restrict__ priors,
                        float* __restrict__ btov, int* __restrict__ btidx,
                        unsigned long long* __restrict__ bestkey) {
  int b = blockIdx.y;
  int p = blockIdx.x * 256 + threadIdx.x;
  __shared__ float sgb[NOBJn * 4];
  __shared__ unsigned long long skey[NOBJn];
  if (threadIdx.x < NOBJn * 4) sgb[threadIdx.x] = gt_boxes[b * NOBJn * 4 + threadIdx.x] * (1.0f / IMGf);
  if (threadIdx.x < NOBJn) skey[threadIdx.x] = 0ull;
  __syncthreads();

  float best = -1.0f; int bidx = 0;
  if (p < PN) {
    float cx = priors[p * 4 + 0], cy = priors[p * 4 + 1];
    float w  = priors[p * 4 + 2], h  = priors[p * 4 + 3];
    float px0 = cx - 0.5f * w, py0 = cy - 0.5f * h;
    float px1 = cx + 0.5f * w, py1 = cy + 0.5f * h;
    float areaP = (px1 - px0) * (py1 - py0);
    for (int g = 0; g < NOBJn; ++g) {
      float x0 = sgb[g * 4 + 0], y0 = sgb[g * 4 + 1];
      float x1 = sgb[g * 4 + 2], y1 = sgb[g * 4 + 3];
      float ix0 = fmaxf(x0, px0), iy0 = fmaxf(y0, py0);
      float ix1 = fminf(x1, px1), iy1 = fminf(y1, py1);
      float iw = fmaxf(ix1 - ix0, 0.0f), ih = fmaxf(iy1 - iy0, 0.0f);
      float inter = iw * ih;
      float areaA = (x1 - x0) * (y1 - y0);
      float iou = inter / (areaA + areaP - inter);
      if (iou > best) { best = iou; bidx = g; }  // first-max wins (strict >)
      // stable argmax over priors: higher iou wins; tie -> smaller prior idx
      unsigned long long key =
          ((unsigned long long)__float_as_uint(iou) << 32) |
          (unsigned long long)(0xFFFFFFFFu - (unsigned)p);
      atomicMax(&skey[g], key);  // LDS ds_max_u64
    }
    btov[(size_t)b * PN + p] = best;
    btidx[(size_t)b * PN + p] = bidx;
  }
  __syncthreads();
  if (threadIdx.x < NOBJn)
    atomicMax(&bestkey[b * NOBJn + threadIdx.x], skey[threadIdx.x]);
}

// ---------------- K2: in-order scatter overrides (last-writer-wins like reference) ----------------
__global__ void k_scatter(const unsigned long long* __restrict__ bestkey,
                          float* __restrict__ btov, int* __restrict__ btidx) {
  int b = threadIdx.x;
  if (b >= BN) return;
  for (int j = 0; j < NOBJn; ++j) {
    unsigned long long key = bestkey[b * NOBJn + j];
    int p = (int)(0xFFFFFFFFu - (unsigned)(key & 0xFFFFFFFFu));
    btov[(size_t)b * PN + p] = 2.0f;
    btidx[(size_t)b * PN + p] = j;
  }
}

// ---------------- K3: heavy streaming kernel ----------------
// conf_data slab (16 priors x 91 classes, contiguous) is staged via
// GLOBAL_LOAD_ASYNC_TO_LDS_B128 (ASYNCcnt path), then reduced with
// V_WMMA_F32_16X16X4_F32 (exact f32 sum of exp).
#define TILE 16
#define WAVES 8
__global__ void k_ce(const float* __restrict__ loc_data, const float* __restrict__ conf_data,
                     const float* __restrict__ gt_boxes, const int* __restrict__ gt_labels,
                     const float* __restrict__ priors,
                     const float* __restrict__ btov, const int* __restrict__ btidx,
                     float* __restrict__ mine, int* __restrict__ numpos,
                     float* __restrict__ scal /* [0]=loss_l, [1]=pos_ce */) {
  __shared__ float sconf[WAVES * TILE * NC];  // 8 * 1456 floats = 46592 B (flat slabs)
  int b = blockIdx.y;
  int wave = threadIdx.x >> 5;
  int lane = threadIdx.x & 31;
  int tile0 = blockIdx.x * (WAVES * TILE) + wave * TILE;
  float* slab = &sconf[wave * TILE * NC];

  // prefetch next block's slab (hint only)
  {
    int pf = tile0 + WAVES * TILE; if (pf > PN - 1) pf = PN - 1;
    __builtin_prefetch(conf_data + ((size_t)b * PN + pf) * NC, 0, 1);
  }

  // async-stage the contiguous 16x91 f32 slab (5824 B) as B128 per-lane transfers:
  // 364 float4's -> 12 predicated rounds of 32 lanes, 512 B per instruction.
  {
    const size_t bBase = (size_t)b * (size_t)PN * NC;
    const int slabVec4 = (TILE * NC) / 4;            // 364
    const long gElemMax = (long)PN * NC - 4;         // in-bounds clamp for tail tiles
    for (int it = 0; it < (slabVec4 + 31) / 32; ++it) {
      int idx4 = it * 32 + lane;
      if (idx4 < slabVec4) {
        long ge = (long)tile0 * NC + (long)idx4 * 4;
        if (ge > gElemMax) ge = gElemMax;            // stays 16B-aligned
        as1_v4i* g = (as1_v4i*)(uintptr_t)(conf_data + bBase + (size_t)ge);
        as3_v4i* l = (as3_v4i*)(unsigned)(uintptr_t)(slab + idx4 * 4);
        __builtin_amdgcn_global_load_async_to_lds_b128(g, l, 0, 0);
      }
    }
  }
  asm volatile("s_wait_asynccnt 0x0" ::: "memory");
  __syncthreads();

  // sum of exp over classes via V_WMMA_F32_16X16X4_F32 with all-ones B.
  // A layout: lane holds row M=(lane&15), K = 2*(lane>>4) + {0,1}.
  int row = lane & 15, half = lane >> 4;
  v8f acc = {0.f, 0.f, 0.f, 0.f, 0.f, 0.f, 0.f, 0.f};
  v2f ones = {1.0f, 1.0f};
#pragma unroll
  for (int kk = 0; kk < (NC + 3) / 4; ++kk) {        // 23 chunks cover classes 0..91
    int c0 = kk * 4 + half * 2;
    int base = row * NC + c0;
    v2f a;
    a.x = __expf(slab[base]);
    a.y = (c0 + 1 < NC) ? __expf(slab[base + 1]) : 0.0f;  // zero the single pad element
    acc = __builtin_amdgcn_wmma_f32_16x16x4_f32(false, a, false, ones,
                                                (short)0, acc, false, false);
  }
  // C[m][n] = sumexp[m] replicated over n; lanes 0-15 hold rows 0-7, lanes 16-31 rows 8-15
  union { v8f v; float f[8]; } u; u.v = acc;
  int m = half * 8 + (lane & 7);
  bool active = (lane & 15) < 8;
  int p = tile0 + m;
  if (active && p < PN) {
    float sumexp = u.f[lane & 7];
    float lse = __logf(sumexp);
    size_t bp = (size_t)b * PN + p;
    int ti = btidx[bp];
    float ov = btov[bp];
    int lab = gt_labels[b * NOBJn + ti];
    int cls = (ov < IOUTH) ? 0 : lab;
    float gathered = slab[m * NC + cls];
    float ce = lse - gathered;
    bool pos = cls > 0;
    mine[bp] = pos ? 0.0f : ce;
    if (pos) {
      atomicAdd(&numpos[b], 1);
      atomicAdd(&scal[1], ce);
      // encode matched GT box vs prior, SmoothL1 vs loc_data
      const float* gb = gt_boxes + (size_t)(b * NOBJn + ti) * 4;
      float x0 = gb[0] * (1.0f / IMGf), y0 = gb[1] * (1.0f / IMGf);
      float x1 = gb[2] * (1.0f / IMGf), y1 = gb[3] * (1.0f / IMGf);
      float cx = priors[p * 4 + 0], cy = priors[p * 4 + 1];
      float w  = priors[p * 4 + 2], h  = priors[p * 4 + 3];
      float t0 = ((x0 + x1) * 0.5f - cx) / (0.1f * w);
      float t1 = ((y0 + y1) * 0.5f - cy) / (0.1f * h);
      float t2 = __logf((x1 - x0) / w) * 5.0f;
      float t3 = __logf((y1 - y0) / h) * 5.0f;
      float t[4] = {t0, t1, t2, t3};
      const float* ld = loc_data + bp * 4;
      float s = 0.0f;
      for (int d = 0; d < 4; ++d) {
        float diff = ld[d] - t[d];
        float ad = fabsf(diff);
        s += (ad < 1.0f) ? 0.5f * diff * diff : ad - 0.5f;
      }
      atomicAdd(&scal[0], s);
    }
  }
}

// ---------------- K4: per-batch exact top-k negative sum via radix-select ----------------
__global__ void k_select(const float* __restrict__ mine, const int* __restrict__ numpos,
                         float* __restrict__ pbneg) {
  int b = blockIdx.x;
  int tid = threadIdx.x;  // 256 threads
  __shared__ unsigned int hist[256];
  __shared__ unsigned int sPref;
  __shared__ int sK;
  __shared__ float sred[256];
  __shared__ int cred[256];

  int np = numpos[b];
  int k = 3 * np; if (k < 1) k = 1; if (k > PN - 1) k = PN - 1;
  unsigned int pref = 0; int kk = k;
  const float* mb = mine + (size_t)b * PN;

  for (int pass = 0; pass < 4; ++pass) {
    int shift = 24 - pass * 8;
    unsigned int mask = (pass == 0) ? 0u : (0xFFFFFFFFu << (shift + 8));
    hist[tid] = 0;
    __syncthreads();
    for (int i = tid; i < PN; i += 256) {
      unsigned int v = __float_as_uint(mb[i]);  // mine >= 0 -> uint order == float order
      if ((v & mask) == pref) atomicAdd(&hist[(v >> shift) & 0xFFu], 1u);
    }
    __syncthreads();
    if (tid == 0) {
      int cum = 0; unsigned int sel = 0; int krem = kk;
      for (int bkt = 255; bkt >= 0; --bkt) {
        int c = (int)hist[bkt];
        if (cum + c >= krem) { sel = (unsigned)bkt; krem -= cum; break; }
        cum += c;
      }
      sPref = pref | (sel << shift);
      sK = krem;
    }
    __syncthreads();
    pref = sPref; kk = sK;
    __syncthreads();
  }
  float T = __uint_as_float(pref);  // exact k-th largest value

  float s = 0.0f; int c = 0;
  for (int i = tid; i < PN; i += 256) {
    float v = mb[i];
    if (v > T) { s += v; c += 1; }
  }
  sred[tid] = s; cred[tid] = c;
  __syncthreads();
  for (int off = 128; off > 0; off >>= 1) {
    if (tid < off) { sred[tid] += sred[tid + off]; cred[tid] += cred[tid + off]; }
    __syncthreads();
  }
  if (tid == 0) pbneg[b] = sred[0] + (float)(k - cred[0]) * T;
}

// ---------------- K5: finalize ----------------
__global__ void k_final(const int* __restrict__ numpos, const float* __restrict__ pbneg,
                        const float* __restrict__ scal, float* __restrict__ out) {
  __shared__ int sn[BN];
  __shared__ float sf[BN];
  int t = threadIdx.x;
  if (t < BN) { sn[t] = numpos[t]; sf[t] = pbneg[t]; }
  __syncthreads();
  if (t == 0) {
    int tot = 0; float neg = 0.0f;
    for (int i = 0; i < BN; ++i) { tot += sn[i]; neg += sf[i]; }
    float N = fmaxf((float)tot, 1.0f);
    out[0] = scal[0] / N;
    out[1] = (scal[1] + neg) / N;
  }
}

extern "C" void kernel_launch(void* const* d_in, const int* in_sizes, int n_in,
                              void* d_out, int out_size, void* d_ws, size_t ws_size,
                              hipStream_t stream) {
  const float* loc_data  = (const float*)d_in[0];
  const float* conf_data = (const float*)d_in[1];
  const float* gt_boxes  = (const float*)d_in[2];
  const int*   gt_labels = (const int*)d_in[3];
  const float* priors    = (const float*)d_in[4];
  float* out = (float*)d_out;

  const size_t BP4 = (size_t)BN * PN * sizeof(float);  // 2,235,392 B
  char* ws = (char*)d_ws;
  unsigned long long* bestkey = (unsigned long long*)ws;                 // 8 KB
  float* btov   = (float*)(ws + 8192);
  int*   btidx  = (int*)  (ws + 8192 + BP4);
  float* mine   = (float*)(ws + 8192 + 2 * BP4);
  int*   numpos = (int*)  (ws + 8192 + 3 * BP4);
  float* pbneg  = (float*)(ws + 8192 + 3 * BP4 + 256);
  float* scal   = (float*)(ws + 8192 + 3 * BP4 + 512);

  k_init<<<4, 256, 0, stream>>>(bestkey, numpos, scal);

  dim3 g1((PN + 255) / 256, BN);
  k_match<<<g1, 256, 0, stream>>>(gt_boxes, priors, btov, btidx, bestkey);

  k_scatter<<<1, 64, 0, stream>>>(bestkey, btov, btidx);

  dim3 g3((PN + WAVES * TILE - 1) / (WAVES * TILE), BN);
  k_ce<<<g3, 256, 0, stream>>>(loc_data, conf_data, gt_boxes, gt_labels, priors,
                               btov, btidx, mine, numpos, scal);

  k_select<<<BN, 256, 0, stream>>>(mine, numpos, pbneg);

  k_final<<<1, 64, 0, stream>>>(numpos, pbneg, scal, out);
}